// CollisionDistanceEvaluator_7456063226430
// MI455X (gfx1250) — compile-verified
//
#include <hip/hip_runtime.h>
#include <math.h>

typedef __attribute__((ext_vector_type(2))) float v2f;
typedef __attribute__((ext_vector_type(8))) float v8f;

#define DIST_COEFF 10000.0f

// Box constants (computed in double, rounded to f32 like numpy would)
#define BCX (-0.001782f)
#define BCY (1.005e-05f)
#define BCZ (0.0431621f)
#define HEX ((float)(0.204416 / 2.0 + 0.001))
#define HEY ((float)(0.0632517 / 2.0 + 0.001))
#define HEZ ((float)(0.1381738 / 2.0 + 0.001))

__global__ __launch_bounds__(256) void collision_dist_kernel(
    const float* __restrict__ trans,
    const float* __restrict__ quat,
    const float* __restrict__ pc,
    float* __restrict__ out,
    int N)
{
    const int b      = blockIdx.x;
    const int tid    = threadIdx.x;
    const int lane   = tid & 31;
    const int wave   = tid >> 5;
    const int nwaves = blockDim.x >> 5;
    const int m      = lane & 15;
    const bool hi    = lane >= 16;

    // ---- per-block scalars: build affine transform from inverse quaternion ----
    const float qx = quat[b * 4 + 0], qy = quat[b * 4 + 1];
    const float qz = quat[b * 4 + 2], qw = quat[b * 4 + 3];
    const float tx = trans[b * 3 + 0], ty = trans[b * 3 + 1], tz = trans[b * 3 + 2];

    const float s2 = qx * qx + qy * qy + qz * qz + qw * qw;
    // qi = conj(q)/|q|^2 ; result = qi * p * conj(qi)
    const float ux = -qx / s2, uy = -qy / s2, uz = -qz / s2, uw = qw / s2;
    const float tt = uw * uw - (ux * ux + uy * uy + uz * uz);

    const float M00 = tt + 2.f * ux * ux;
    const float M01 = 2.f * (ux * uy - uw * uz);
    const float M02 = 2.f * (ux * uz + uw * uy);
    const float M10 = 2.f * (ux * uy + uw * uz);
    const float M11 = tt + 2.f * uy * uy;
    const float M12 = 2.f * (uy * uz - uw * ux);
    const float M20 = 2.f * (ux * uz - uw * uy);
    const float M21 = 2.f * (uy * uz + uw * ux);
    const float M22 = tt + 2.f * uz * uz;

    // Per-lane A-matrix slots (16x4 f32 layout, wave32):
    //   lanes 0-15 : VGPR0 = A[m][0], VGPR1 = A[m][1]
    //   lanes16-31 : VGPR0 = A[m][2], VGPR1 = A[m][3]
    // Rows 0..2 = [M_row | -trans], rows 3..15 = 0.
    float c0 = 0.f, c1 = 0.f, c2 = 0.f, c3 = 0.f;
    if (m == 0)      { c0 = M00; c1 = M01; c2 = M02; c3 = -tx; }
    else if (m == 1) { c0 = M10; c1 = M11; c2 = M12; c3 = -ty; }
    else if (m == 2) { c0 = M20; c1 = M21; c2 = M22; c3 = -tz; }
    v2f A;
    A.x = hi ? c2 : c0;
    A.y = hi ? c3 : c1;

    // AABB bounds
    const float lox = BCX - HEX, hix = BCX + HEX;
    const float loy = BCY - HEY, hiy = BCY + HEY;
    const float loz = BCZ - HEZ, hiz = BCZ + HEZ;

    // B-matrix (4x16 f32, wave32) load offsets within a 16-point (192B) group:
    //   VGPR0: lanes0-15 -> x (=3n+0), lanes16-31 -> z (=3n+2)
    //   VGPR1: lanes0-15 -> y (=3n+1), lanes16-31 -> constant 1.0
    const int r0 = m * 12 + (hi ? 8 : 0);
    const int r1 = hi ? 0 : (m * 12 + 4); // hi lanes: dummy in-bounds load, overwritten
    const int idxb = m * 4;               // ds_bpermute byte index: hi lane n+16 <- lane n

    // Unconditional prefetch: lanes clamped to 3 distinct 128B lines (no EXEC churn)
    const int poff = (lane < 2 ? lane : 2) * 128;

    const char* rb = (const char*)pc + (size_t)b * (size_t)N * 12u;

    float accS = 0.f, accC = 0.f;
    const v8f cacc = {0.f, 0.f, 0.f, 0.f, 0.f, 0.f, 0.f, 0.f};

    const int nch = N >> 5; // chunks of 32 points (384 bytes)
    #pragma unroll 2
    for (int ch = wave; ch < nch; ch += nwaves) {
        const size_t base = (size_t)ch * 384u;

        // Prefetch this wave's chunk one stride ahead (8 chunks = 3072B)
        __builtin_prefetch(rb + base + 3072 + poff, 0, 1);

        // Points ch*32 .. ch*32+15
        v2f B1;
        B1.x = *(const float*)(rb + base + r0);
        {
            const float yv = *(const float*)(rb + base + r1);
            B1.y = hi ? 1.0f : yv;
        }
        // Points ch*32+16 .. ch*32+31
        v2f B2;
        B2.x = *(const float*)(rb + base + 192 + r0);
        {
            const float yv = *(const float*)(rb + base + 192 + r1);
            B2.y = hi ? 1.0f : yv;
        }

        // D rows 0..2 (lanes 0-15) = transformed x',y',z'
        v8f d1 = __builtin_amdgcn_wmma_f32_16x16x4_f32(
            false, A, false, B1, (short)0, cacc, false, false);
        v8f d2 = __builtin_amdgcn_wmma_f32_16x16x4_f32(
            false, A, false, B2, (short)0, cacc, false, false);

        // Move d2 results (lanes 0-15) into lanes 16-31 so all 32 lanes work.
        const float bx = __int_as_float(
            __builtin_amdgcn_ds_bpermute(idxb, __float_as_int(d2[0])));
        const float by = __int_as_float(
            __builtin_amdgcn_ds_bpermute(idxb, __float_as_int(d2[1])));
        const float bz = __int_as_float(
            __builtin_amdgcn_ds_bpermute(idxb, __float_as_int(d2[2])));

        const float px = hi ? bx : d1[0];
        const float py = hi ? by : d1[1];
        const float pz = hi ? bz : d1[2];

        const bool in =
            (px >= lox) & (px <= hix) &
            (py >= loy) & (py <= hiy) &
            (pz >= loz) & (pz <= hiz);
        // Raw hardware sqrt (v_sqrt_f32, ~1 ulp) — avoids the precise-sqrt expansion
        const float nr = __builtin_amdgcn_sqrtf(px * px + py * py + pz * pz);
        accS += in ? nr : 0.0f;
        accC += in ? 1.0f : 0.0f;
    }

    // ---- wave reduction (wave32) ----
    for (int off = 16; off > 0; off >>= 1) {
        accS += __shfl_xor(accS, off, 32);
        accC += __shfl_xor(accC, off, 32);
    }

    __shared__ float sS[32];
    __shared__ float sC[32];
    if (lane == 0) { sS[wave] = accS; sC[wave] = accC; }
    __syncthreads();

    if (tid == 0) {
        float S = 0.f, C = 0.f;
        for (int i = 0; i < nwaves; ++i) { S += sS[i]; C += sC[i]; }
        out[b] = (C > 0.f) ? (-S / C) * DIST_COEFF : DIST_COEFF;
    }
}

extern "C" void kernel_launch(void* const* d_in, const int* in_sizes, int n_in,
                              void* d_out, int out_size, void* d_ws, size_t ws_size,
                              hipStream_t stream) {
    const float* trans = (const float*)d_in[0]; // (B,3)
    const float* quat  = (const float*)d_in[1]; // (B,4)
    const float* pc    = (const float*)d_in[2]; // (B,N,3)

    const int B = in_sizes[0] / 3;
    const int N = (int)(in_sizes[2] / ((long long)B * 3));

    collision_dist_kernel<<<dim3(B), dim3(256), 0, stream>>>(
        trans, quat, pc, (float*)d_out, N);
}